// DistancePreprocessingLayer_70824010711584
// MI455X (gfx1250) — compile-verified
//
#include <hip/hip_runtime.h>
#include <hip/hip_bf16.h>

typedef __attribute__((ext_vector_type(2))) float v2f;
typedef __attribute__((ext_vector_type(4))) float v4f;
typedef __attribute__((ext_vector_type(8))) float v8f;

#define N_NODES 512
#define DIM     64
#define NPAIRS  (N_NODES * (N_NODES - 1) / 2)   // 130816 = 511*256
#define P4      (NPAIRS / 4)                    // 32704 float4s per row

// ---------------------------------------------------------------------------
// Stage 0: squared norms of each row of z  (512 values, trivial)
// ---------------------------------------------------------------------------
__global__ void norms_kernel(const float* __restrict__ z, float* __restrict__ norms) {
    int i = blockIdx.x * blockDim.x + threadIdx.x;
    if (i < N_NODES) {
        const v4f* row = (const v4f*)(z + i * DIM);
        float s = 0.0f;
#pragma unroll
        for (int k = 0; k < DIM / 4; ++k) {
            v4f v = row[k];
            s += v.x * v.x + v.y * v.y + v.z * v.z + v.w * v.w;
        }
        norms[i] = s;
    }
}

// ---------------------------------------------------------------------------
// Stage 1: Gram tiles via V_WMMA_F32_16X16X4_F32, convert to packed triu
// distances. One wave per 16x16 tile; only tj >= ti tiles do work.
//   A 16x4 tile: lane -> m = lane&15, k-pair = 2*(lane>>4)   (2 VGPRs = v2f)
//   B  4x16 tile: lane -> n = lane&15, k-pair = 2*(lane>>4)  (row read of Z)
//   C 16x16:     vgpr r, lane -> (m = r + 8*(lane>>4), n = lane&15)
// ---------------------------------------------------------------------------
__global__ void __launch_bounds__(32)
gram_dist_kernel(const float* __restrict__ z,
                 const float* __restrict__ norms,
                 float* __restrict__ dist) {
    const int ti = blockIdx.x;
    const int tj = blockIdx.y;
    if (tj < ti) return;                       // lower-triangular tiles: no pairs

    const int lane  = threadIdx.x;             // wave32, EXEC all ones here
    const int half  = lane >> 4;               // 0: K={0,1} / 1: K={2,3} stripe
    const int l15   = lane & 15;
    const int ibase = ti * 16;
    const int jbase = tj * 16;

    const float* arow = z + (ibase + l15) * DIM + 2 * half;  // A row m = l15
    const float* brow = z + (jbase + l15) * DIM + 2 * half;  // B col n = l15

    v8f c = {};
#pragma unroll
    for (int kk = 0; kk < DIM; kk += 4) {
        v2f a = *(const v2f*)(arow + kk);
        v2f b = *(const v2f*)(brow + kk);
        // 8 args: (neg_a, A, neg_b, B, c_mod, C, reuse_a, reuse_b)
        c = __builtin_amdgcn_wmma_f32_16x16x4_f32(false, a, false, b,
                                                  (short)0, c, false, false);
    }

    // d(i,j) = sqrt(max(0, ni + nj - 2*gij)); pack at triu row-major index.
    const int   j  = jbase + l15;
    const float nj = norms[j];
#pragma unroll
    for (int r = 0; r < 8; ++r) {
        const int i = ibase + r + 8 * half;
        if (i < j) {
            float d2 = norms[i] + nj - 2.0f * c[r];
            float dd = sqrtf(fmaxf(d2, 0.0f));
            int   p  = i * (2 * N_NODES - i - 1) / 2 + (j - i - 1);
            dist[p] = dd;
        }
    }
}

// ---------------------------------------------------------------------------
// Stage 2: broadcast dist[P] to all 512 output rows. This is the whole
// roofline: 268 MB of stores (~11.5 us @ 23.3 TB/s). Each thread loads one
// float4 of dist once, then streams it to 64 rows with non-temporal 128-bit
// stores (output >> L2, keep it out of cache).
// ---------------------------------------------------------------------------
__global__ void __launch_bounds__(256)
broadcast_kernel(const float* __restrict__ dist, float* __restrict__ out) {
    const int idx4 = blockIdx.x * blockDim.x + threadIdx.x;
    if (idx4 >= P4) return;

    const v4f v = ((const v4f*)dist)[idx4];

    const int rowsPer = N_NODES / 8;           // gridDim.y == 8 -> 64 rows
    const int r0      = blockIdx.y * rowsPer;
    v4f* o = (v4f*)out + (long)r0 * P4 + idx4;
#pragma unroll 4
    for (int r = 0; r < rowsPer; ++r) {
        __builtin_nontemporal_store(v, o);     // global_store_b128 ... nt
        o += P4;
    }
}

// ---------------------------------------------------------------------------
extern "C" void kernel_launch(void* const* d_in, const int* in_sizes, int n_in,
                              void* d_out, int out_size, void* d_ws, size_t ws_size,
                              hipStream_t stream) {
    const float* z   = (const float*)d_in[0];
    float*       out = (float*)d_out;

    // scratch layout: [0..511] row norms, [512 .. 512+NPAIRS) packed distances
    float* norms = (float*)d_ws;
    float* dist  = norms + N_NODES;

    norms_kernel<<<dim3(2), dim3(256), 0, stream>>>(z, norms);

    gram_dist_kernel<<<dim3(N_NODES / 16, N_NODES / 16), dim3(32), 0, stream>>>(
        z, norms, dist);

    broadcast_kernel<<<dim3((P4 + 255) / 256, 8), dim3(256), 0, stream>>>(dist, out);
}